// CasRel_23021024707241
// MI455X (gfx1250) — compile-verified
//
#include <hip/hip_runtime.h>
#include <math.h>
#include <stdint.h>

#define Bsz 64
#define Ssz 512
#define Hsz 768
#define Rsz 51
#define Gsz 3072     // 4*H
#define Tsz 16       // MAX_SPAN
#define LSTRIDE 772  // LDS row stride in floats (768 + 4 pad -> conflict-free column reads)
#define EPSF 1e-12f

typedef __attribute__((ext_vector_type(2))) float v2f;
typedef __attribute__((ext_vector_type(8))) float v8f;

// D = A(16x4, f32) * B(4x16, f32) + C(16x16, f32), full wave32 op
__device__ __forceinline__ v8f wmma4(v2f a, v2f b, v8f c) {
  return __builtin_amdgcn_wmma_f32_16x16x4_f32(false, a, false, b, (short)0, c,
                                               false, false);
}

__device__ __forceinline__ float sigf(float x) { return 1.0f / (1.0f + __expf(-x)); }

// CDNA5 async copy: 16 bytes per lane, global -> LDS, no VGPR round trip.
// GV addressing form; tracked by ASYNCcnt. lds_addr = wave-relative LDS byte addr
// (low 32 bits of the generic pointer to __shared__), gaddr = 64-bit global addr.
__device__ __forceinline__ void async_ld16(uint32_t lds_addr, const void* gptr) {
  asm volatile("global_load_async_to_lds_b128 %0, %1, off"
               :
               : "v"(lds_addr), "v"((unsigned long long)(uintptr_t)gptr)
               : "memory");
}
__device__ __forceinline__ void async_wait0() {
  asm volatile("s_wait_asynccnt 0x0" ::: "memory");
}
__device__ __forceinline__ uint32_t lds_addr_of(const void* shared_ptr) {
  return (uint32_t)(uintptr_t)shared_ptr;  // addr[31:0] == LDS byte offset
}

__global__ void k_zero(float* __restrict__ p, int n) {
  int i = blockIdx.x * blockDim.x + threadIdx.x;
  if (i < n) p[i] = 0.0f;
}

// ---------------------------------------------------------------------------
// Kernel A: xW[b,t,:] = embed[b, clip(head[b]+t), :] @ W_ih^T + b_ih + b_hh
// M = B*T = 1024 rows (one M-tile == one batch, rows are t=0..15)
// grid (64, 12), block 512 (16 waves); wave w owns N-tile n0 = by*256 + w*16
// ---------------------------------------------------------------------------
__global__ void k_xw(const float* __restrict__ embed, const int* __restrict__ head,
                     const float* __restrict__ Wih, const float* __restrict__ bih,
                     const float* __restrict__ bhh, float* __restrict__ xw) {
  __shared__ float xs[16 * LSTRIDE];
  const int b   = blockIdx.x;
  const int tid = threadIdx.x;
  const int hd  = head[b];
  // stage gathered A tile (16 rows x 768) into LDS via async-to-LDS, 16B chunks
  {
    const uint32_t lbase = lds_addr_of(xs);
    for (int e = tid * 4; e < 16 * Hsz; e += 512 * 4) {
      int t = e / Hsz, h = e - t * Hsz;  // 768 % 4 == 0 -> chunk stays in-row
      int pos = hd + t;
      pos = pos < 0 ? 0 : (pos > Ssz - 1 ? Ssz - 1 : pos);
      async_ld16(lbase + (uint32_t)(t * LSTRIDE + h) * 4u,
                 &embed[((size_t)b * Ssz + pos) * Hsz + h]);
    }
    async_wait0();
  }
  __syncthreads();

  const int wave = tid >> 5;
  const int lane = tid & 31;
  const int l15  = lane & 15;
  const int hi   = lane >> 4;
  const int n    = blockIdx.y * 256 + wave * 16 + l15;
  const float* wrow = Wih + (size_t)n * Hsz;

  v8f c = {0.f, 0.f, 0.f, 0.f, 0.f, 0.f, 0.f, 0.f};
  for (int k0 = 0; k0 < Hsz; k0 += 4) {
    int kb = k0 + (hi << 1);
    v2f a  = *(const v2f*)&xs[l15 * LSTRIDE + kb];   // A[M=l15][K=kb..kb+1]
    v2f bb = *(const v2f*)&wrow[kb];                 // B[K][N=n] = W_ih[n][K]
    c = wmma4(a, bb, c);
  }
  float bias = bih[n] + bhh[n];
#pragma unroll
  for (int v = 0; v < 8; v++) {
    int t = v + (hi << 3);  // row within tile
    xw[((size_t)b * Tsz + t) * Gsz + n] = c[v] + bias;
  }
}

// ---------------------------------------------------------------------------
// Kernel B (x16, sequential): gates = xW[:,t,:] + h_in @ W_hh^T; LSTM cell.
// grid (4, 48), block 128 (4 waves). WG owns 16 batches x 16 hidden units;
// wave g computes gate g's 16x16 tile (C seeded from xW), then fused cell.
// h double-buffered across step kernels; c updated in place (WG-exclusive).
// W_hh (9.4 MB) and h (196 KB) stay resident in the 192 MB L2 across steps.
// ---------------------------------------------------------------------------
__global__ void k_lstm_step(const float* __restrict__ xw, const float* __restrict__ Whh,
                            const float* __restrict__ hin, float* __restrict__ hout,
                            float* __restrict__ cbuf, const int* __restrict__ head,
                            const int* __restrict__ tail, int t) {
  __shared__ float hs[16 * LSTRIDE];
  __shared__ float gb[4 * 16 * 16];
  const int b0  = blockIdx.x * 16;
  const int h0  = blockIdx.y * 16;
  const int tid = threadIdx.x;
  {
    const uint32_t lbase = lds_addr_of(hs);
    for (int e = tid * 4; e < 16 * Hsz; e += 128 * 4) {
      int m = e / Hsz, h = e - m * Hsz;
      async_ld16(lbase + (uint32_t)(m * LSTRIDE + h) * 4u,
                 &hin[(size_t)(b0 + m) * Hsz + h]);
    }
    async_wait0();
  }
  __syncthreads();

  const int g    = tid >> 5;  // gate index == wave
  const int lane = tid & 31;
  const int l15  = lane & 15;
  const int hi   = lane >> 4;
  const int n    = g * Hsz + h0 + l15;
  const float* wrow = Whh + (size_t)n * Hsz;

  v8f c;
#pragma unroll
  for (int v = 0; v < 8; v++) {
    int m = v + (hi << 3);
    c[v]  = xw[((size_t)(b0 + m) * Tsz + t) * Gsz + n];  // seed C with x-part
  }
  for (int k0 = 0; k0 < Hsz; k0 += 4) {
    int kb = k0 + (hi << 1);
    v2f a  = *(const v2f*)&hs[l15 * LSTRIDE + kb];
    v2f bb = *(const v2f*)&wrow[kb];
    c = wmma4(a, bb, c);
  }
#pragma unroll
  for (int v = 0; v < 8; v++) {
    int m = v + (hi << 3);
    gb[(g * 16 + m) * 16 + l15] = c[v];
  }
  __syncthreads();

  for (int cidx = tid; cidx < 256; cidx += 128) {
    int m = cidx >> 4, nn = cidx & 15;
    float gi = gb[(0 * 16 + m) * 16 + nn];
    float gf = gb[(1 * 16 + m) * 16 + nn];
    float gg = gb[(2 * 16 + m) * 16 + nn];
    float go = gb[(3 * 16 + m) * 16 + nn];
    size_t idx  = (size_t)(b0 + m) * Hsz + h0 + nn;
    float c_old = cbuf[idx];
    float h_old = hs[m * LSTRIDE + h0 + nn];
    float c_new = sigf(gf) * c_old + sigf(gi) * tanhf(gg);
    float h_new = sigf(go) * tanhf(c_new);
    bool valid  = (head[b0 + m] + t) <= tail[b0 + m];
    cbuf[idx] = valid ? c_new : c_old;
    hout[idx] = valid ? h_new : h_old;
  }
}

// ---------------------------------------------------------------------------
// Kernel C: wln = hn @ Ww^T + cln_weight ; bln = hn @ Wb^T + cln_bias
// grid (4, 48, 2), block 32 (one wave -> one 16x16 tile, K=768)
// ---------------------------------------------------------------------------
__global__ void k_cln(const float* __restrict__ hn, const float* __restrict__ Ww,
                      const float* __restrict__ Wb, const float* __restrict__ wconst,
                      const float* __restrict__ bconst, float* __restrict__ wln,
                      float* __restrict__ bln) {
  const int m0  = blockIdx.x * 16;
  const int n0  = blockIdx.y * 16;
  const int z   = blockIdx.z;
  const int lane = threadIdx.x & 31;
  const int l15  = lane & 15;
  const int hi   = lane >> 4;
  const float* W    = z ? Wb : Ww;
  const float* wrow = W + (size_t)(n0 + l15) * Hsz;
  const float* arow = hn + (size_t)(m0 + l15) * Hsz;

  v8f c = {0.f, 0.f, 0.f, 0.f, 0.f, 0.f, 0.f, 0.f};
  for (int k0 = 0; k0 < Hsz; k0 += 4) {
    int kb = k0 + (hi << 1);
    v2f a  = *(const v2f*)&arow[kb];
    v2f bb = *(const v2f*)&wrow[kb];
    c = wmma4(a, bb, c);
  }
  float add  = (z ? bconst : wconst)[n0 + l15];
  float* out = z ? bln : wln;
#pragma unroll
  for (int v = 0; v < 8; v++) {
    int m = m0 + v + (hi << 3);
    out[(size_t)m * Hsz + n0 + l15] = c[v] + add;
  }
}

// ---------------------------------------------------------------------------
// Kernel D: fused LayerNorm + dual projection + sigmoid.
// grid 2048 (16 rows of (b,s) each), block 256 (8 waves).
// Phase 0: async-to-LDS stage of 16x768 embed rows (the 100 MB stream).
// Phase 1: wave shfl_xor reductions for mean/var, normalize in LDS.
// Phase 2: waves 0..6 each do one 16x16 N-tile of the 16x102 output GEMM.
// new_embed (100 MB) never hits HBM.
// ---------------------------------------------------------------------------
__global__ void k_ln_proj(const float* __restrict__ embed, const float* __restrict__ wln,
                          const float* __restrict__ bln, const float* __restrict__ Wh,
                          const float* __restrict__ bh, const float* __restrict__ Wt,
                          const float* __restrict__ bt, float* __restrict__ out) {
  __shared__ float xs[16 * LSTRIDE];
  __shared__ float mu_s[16], ri_s[16];
  const int r0  = blockIdx.x * 16;
  const int b   = r0 / Ssz;           // 512 % 16 == 0 -> tile stays in one batch
  const int s0  = r0 - b * Ssz;
  const int tid = threadIdx.x;

  {
    const uint32_t lbase = lds_addr_of(xs);
    for (int e = tid * 4; e < 16 * Hsz; e += 256 * 4) {
      int i = e / Hsz, h = e - i * Hsz;
      async_ld16(lbase + (uint32_t)(i * LSTRIDE + h) * 4u,
                 &embed[((size_t)b * Ssz + s0 + i) * Hsz + h]);
    }
    async_wait0();
  }
  __syncthreads();

  const int wave = tid >> 5, lane = tid & 31;
  const int l15 = lane & 15, hig = lane >> 4;
  {
    int row = wave * 2 + hig;  // 8 waves x 2 rows = 16 rows, 16 lanes/row
    float s = 0.f, sq = 0.f;
    for (int h = l15; h < Hsz; h += 16) {
      float x = xs[row * LSTRIDE + h];
      s += x;
      sq += x * x;
    }
    for (int m = 1; m <= 8; m <<= 1) {  // butterfly within 16-lane half
      s  += __shfl_xor(s, m, 32);
      sq += __shfl_xor(sq, m, 32);
    }
    float mu  = s * (1.0f / Hsz);
    float var = sq * (1.0f / Hsz) - mu * mu;
    float ri  = 1.0f / (sqrtf(var + EPSF) + EPSF);
    if (l15 == 0) { mu_s[row] = mu; ri_s[row] = ri; }
  }
  __syncthreads();

  for (int e = tid; e < 16 * Hsz; e += 256) {
    int i = e / Hsz, h = e - i * Hsz;
    float x = xs[i * LSTRIDE + h];
    xs[i * LSTRIDE + h] =
        wln[(size_t)b * Hsz + h] * (x - mu_s[i]) * ri_s[i] + bln[(size_t)b * Hsz + h];
  }
  __syncthreads();

  if (wave < 7) {  // 7 tiles cover 102 output cols (heads 0..50, tails 51..101)
    const int n = wave * 16 + l15;
    const float* wrow = nullptr;
    if (n < Rsz)
      wrow = Wh + (size_t)n * Hsz;
    else if (n < 2 * Rsz)
      wrow = Wt + (size_t)(n - Rsz) * Hsz;

    v8f c = {0.f, 0.f, 0.f, 0.f, 0.f, 0.f, 0.f, 0.f};
    for (int k0 = 0; k0 < Hsz; k0 += 4) {
      int kb = k0 + (hig << 1);
      v2f a  = *(const v2f*)&xs[l15 * LSTRIDE + kb];
      v2f bb;
      if (wrow) {
        bb.x = wrow[kb];
        bb.y = wrow[kb + 1];
      } else {
        bb.x = 0.f;
        bb.y = 0.f;
      }
      c = wmma4(a, bb, c);
    }
    if (n < 2 * Rsz) {
      float bias  = (n < Rsz) ? bh[n] : bt[n - Rsz];
      size_t base = (n < Rsz) ? 0 : (size_t)Bsz * Ssz * Rsz;
      int col     = (n < Rsz) ? n : n - Rsz;
#pragma unroll
      for (int v = 0; v < 8; v++) {
        int row = v + (hig << 3);
        out[base + ((size_t)b * Ssz + s0 + row) * Rsz + col] = sigf(c[v] + bias);
      }
    }
  }
}

extern "C" void kernel_launch(void* const* d_in, const int* in_sizes, int n_in,
                              void* d_out, int out_size, void* d_ws, size_t ws_size,
                              hipStream_t stream) {
  (void)in_sizes; (void)n_in; (void)out_size; (void)ws_size;
  const float* embed = (const float*)d_in[0];
  const int*   head  = (const int*)d_in[1];
  const int*   tail  = (const int*)d_in[2];
  const float* Wih   = (const float*)d_in[3];
  const float* Whh   = (const float*)d_in[4];
  const float* bih   = (const float*)d_in[5];
  const float* bhh   = (const float*)d_in[6];
  const float* clnw  = (const float*)d_in[7];
  const float* clnb  = (const float*)d_in[8];
  const float* Ww    = (const float*)d_in[9];
  const float* Wb    = (const float*)d_in[10];
  const float* Wh    = (const float*)d_in[11];
  const float* bh    = (const float*)d_in[12];
  const float* Wt    = (const float*)d_in[13];
  const float* bt    = (const float*)d_in[14];
  float* out = (float*)d_out;

  float* ws    = (float*)d_ws;
  float* xw    = ws;                              // 64*16*3072
  float* hbuf0 = xw + (size_t)Bsz * Tsz * Gsz;    // 64*768
  float* cbuf  = hbuf0 + (size_t)Bsz * Hsz;       // 64*768 (adjacent for zeroing)
  float* hbuf1 = cbuf + (size_t)Bsz * Hsz;        // 64*768
  float* wln   = hbuf1 + (size_t)Bsz * Hsz;       // 64*768
  float* bln   = wln + (size_t)Bsz * Hsz;         // 64*768

  const int nz = 2 * Bsz * Hsz;  // hbuf0 + cbuf
  k_zero<<<(nz + 255) / 256, 256, 0, stream>>>(hbuf0, nz);

  k_xw<<<dim3(64, 12), 512, 0, stream>>>(embed, head, Wih, bih, bhh, xw);

  for (int t = 0; t < Tsz; t++) {
    const float* hin = (t & 1) ? hbuf1 : hbuf0;
    float* hout      = (t & 1) ? hbuf0 : hbuf1;
    k_lstm_step<<<dim3(4, 48), 128, 0, stream>>>(xw, Whh, hin, hout, cbuf, head, tail, t);
  }
  // after 16 steps the final h lives in hbuf0
  k_cln<<<dim3(4, 48, 2), 32, 0, stream>>>(hbuf0, Ww, Wb, clnw, clnb, wln, bln);

  k_ln_proj<<<2048, 256, 0, stream>>>(embed, wln, bln, Wh, bh, Wt, bt, out);
}